// SE2MultiSelfAttention_32280974196954
// MI455X (gfx1250) — compile-verified
//
#include <hip/hip_runtime.h>
#include <math.h>

// ---------------------------------------------------------------------------
// SE(2) multi-head self-attention, fp32 WMMA pipeline for gfx1250 (MI455X).
// complex64 tensors are interleaved (re,im) float pairs; a (R,C) complex
// row-major matrix is a (R,2C) real row-major matrix, so real-weight GEMMs run
// directly as WMMA f32 16x16x4 GEMMs with 2C columns.
//
// Bandwidth-oriented blocking (HBM 23.3 TB/s is the limiter, not WMMA rate):
//  - k_proj / k_out: 3 output m-tiles per wave -> B fragment loaded once.
//  - k_attn: E0 slab staged in LDS with global_load_async_to_lds_b64 and
//    shared by 8 j-tiles per block (ASYNCcnt + workgroup barrier).
//  - k_av:  V slab (32KB) staged in LDS the same way, shared by 8 i-tiles.
// ---------------------------------------------------------------------------

typedef float v2f __attribute__((ext_vector_type(2)));
typedef float v8f __attribute__((ext_vector_type(8)));

static __device__ __forceinline__ v8f wmma4(v2f a, v2f b, v8f c) {
  // f32 WMMA supports only C-neg in HW; signs are handled in the loads.
  return __builtin_amdgcn_wmma_f32_16x16x4_f32(false, a, false, b, (short)0, c,
                                               false, false);
}

// Async global -> LDS copy of 8 bytes (one b64 element), gfx1250 ASYNCcnt path.
static __device__ __forceinline__ void async_g2l_b64(uint32_t lds_off,
                                                     const float* gptr) {
  uint64_t ga = (uint64_t)(uintptr_t)gptr;
  asm volatile("global_load_async_to_lds_b64 %0, %1, off"
               :: "v"(lds_off), "v"(ga)
               : "memory");
}
static __device__ __forceinline__ void async_wait0() {
  asm volatile("s_wait_asynccnt 0" ::: "memory");
}

static constexpr int BB = 8, MM = 8, HH = 8, QD = 48, DD = 384, NN = 256;
static constexpr int N2 = NN * 2;  // real-view columns of a complex (.,N) mat

// Workspace layout (float offsets)
static constexpr size_t MNN     = (size_t)MM * NN * NN;
static constexpr size_t POSR    = 0;
static constexpr size_t POSI    = MNN;
static constexpr size_t EOFF    = 2 * MNN;                       // E[3,b,m,h,q,2n]
static constexpr size_t ESZ     = (size_t)3 * BB * MM * HH * QD * NN * 2;
static constexpr size_t SOFF    = EOFF + ESZ;                    // s[b,m,h,i] cplx
static constexpr size_t SSZ     = (size_t)BB * MM * HH * NN * 2;
static constexpr size_t AOFF    = SOFF + SSZ;                    // A_sum cplx
static constexpr size_t ASZ     = (size_t)BB * HH * NN * NN * 2;
static constexpr size_t SMOFF   = AOFF + ASZ;                    // softmax(A) real
static constexpr size_t SMSZ    = (size_t)BB * HH * NN * NN;
static constexpr size_t TOFF    = SMOFF + SMSZ;                  // t[b,m,h,i] cplx
static constexpr size_t TSZ     = (size_t)BB * MM * HH * NN * 2;
static constexpr size_t FLATOFF = TOFF + TSZ;                    // flat[b,m,384,2n]

// ---------------------------------------------------------------------------
// Kernel 0: SE(2) positional encoding pos[m,i,j] = r_ij * exp(i*m*theta_ij)
// ---------------------------------------------------------------------------
__global__ void k_pos(float* __restrict__ ws) {
  int idx = blockIdx.x * blockDim.x + threadIdx.x;
  if (idx >= (int)MNN) return;
  int j = idx & 255;
  int i = (idx >> 8) & 255;
  int m = idx >> 16;
  float dy = (float)(i >> 4) - (float)(j >> 4);
  float dx = (float)(i & 15) - (float)(j & 15);
  float r  = sqrtf(dy * dy + dx * dx) * (1.0f / sqrtf(450.0f));
  float th = atan2f(dx, dy) * (float)m;
  ws[POSR + idx] = r * cosf(th);
  ws[POSI + idx] = r * sinf(th);
}

// ---------------------------------------------------------------------------
// Kernel 1: projection E[p,b,m,h] = emb[p,m,h](48x384) @ x[b,m]-view(384x512)
// One wave per (p,b,m,h,nt): all 3 m-tiles, B fragment loaded once.
// ---------------------------------------------------------------------------
__global__ __launch_bounds__(256) void k_proj(const float* __restrict__ x,
                                              const float* __restrict__ emb,
                                              float* __restrict__ ws) {
  int lane = threadIdx.x & 31, wave = threadIdx.x >> 5;
  int batch = blockIdx.x >> 2;                 // (p,b,m,h): 1536
  int nt    = ((blockIdx.x & 3) << 3) + wave;  // 0..31
  int h = batch % HH; int t = batch / HH;
  int m = t % MM;  t /= MM;
  int b = t % BB;  int p = t / BB;

  const float* A  = emb + (((size_t)p * MM + m) * HH + h) * (size_t)QD * DD;
  const float* Bp = x + ((size_t)b * MM + m) * (size_t)DD * N2 + nt * 16;
  float* C = ws + EOFF
           + ((((size_t)p * BB + b) * MM + m) * HH + h) * (size_t)QD * N2
           + nt * 16;

  int r = lane & 15, kh = (lane >> 4) * 2;
  v8f acc[3] = {{}, {}, {}};
  for (int k = 0; k < DD; k += 4) {
    v2f bf;
    bf.x = Bp[(size_t)(k + kh) * N2 + r];
    bf.y = Bp[(size_t)(k + kh + 1) * N2 + r];
#pragma unroll
    for (int mt = 0; mt < 3; mt++) {
      v2f a;
      a.x = A[(size_t)(mt * 16 + r) * DD + k + kh];
      a.y = A[(size_t)(mt * 16 + r) * DD + k + kh + 1];
      acc[mt] = wmma4(a, bf, acc[mt]);
    }
  }
  int col = lane & 15, rb = (lane >> 4) * 8;
#pragma unroll
  for (int mt = 0; mt < 3; mt++)
    for (int v = 0; v < 8; v++)
      C[(size_t)(mt * 16 + rb + v) * N2 + col] = acc[mt][v];
}

// ---------------------------------------------------------------------------
// Kernel 2: s[b,m,h,i] = sum_q conj(E0[q,i]) * enc0[m,h,q]
// ---------------------------------------------------------------------------
__global__ void k_s(const float* __restrict__ enc, float* __restrict__ ws) {
  int idx = blockIdx.x * blockDim.x + threadIdx.x;  // B*M*H*N
  if (idx >= BB * MM * HH * NN) return;
  int i = idx & 255; int t = idx >> 8;
  int h = t % HH; t /= HH;
  int m = t % MM; int b = t / MM;
  const float* E0 = ws + EOFF
                  + (((size_t)b * MM + m) * HH + h) * (size_t)QD * N2;
  const float* e0 = enc + ((size_t)m * HH + h) * QD;
  float sr = 0.f, si = 0.f;
  for (int q = 0; q < QD; q++) {
    float w = e0[q];
    sr += w * E0[(size_t)q * N2 + i * 2];
    si -= w * E0[(size_t)q * N2 + i * 2 + 1];
  }
  ws[SOFF + 2 * (size_t)idx]     = sr;
  ws[SOFF + 2 * (size_t)idx + 1] = si;
}

// ---------------------------------------------------------------------------
// Kernel 3: A_sum[b,h] = sum_m ( E0^H E1 + s * pos_enc ), complex 256x256.
// Block = 8 waves sharing one i-tile; E0 slab (48x16 cplx = 6KB) staged in LDS
// via async global->LDS, A-fragments read from LDS (ds_load).
// ---------------------------------------------------------------------------
__global__ __launch_bounds__(256) void k_attn(float* __restrict__ ws) {
  __shared__ float e0s[QD * 32];                       // [q][ic*2(+1)]
  int lane = threadIdx.x & 31, wave = threadIdx.x >> 5;
  int bh  = blockIdx.x >> 5;                           // 64 (b,h)
  int rem = blockIdx.x & 31;
  int it  = rem >> 1;                                  // 0..15
  int jt  = ((rem & 1) << 3) + wave;                   // 0..15
  int b = bh >> 3, h = bh & 7;
  int r = lane & 15, kh = (lane >> 4) * 2;
  int j = jt * 16 + r;

  v8f accR = {}, accI = {};
  const float* PR = ws + POSR;
  const float* PI = ws + POSI;

  for (int m = 0; m < MM; m++) {
    const float* E0 = ws + EOFF
        + ((((size_t)0 * BB + b) * MM + m) * HH + h) * (size_t)QD * N2;
    const float* E1 = ws + EOFF
        + ((((size_t)1 * BB + b) * MM + m) * HH + h) * (size_t)QD * N2;
    // Prefetch next-m E1 panel while we compute this one.
    if (m + 1 < MM)
      __builtin_prefetch(E1 + (size_t)HH * QD * N2, 0, 0);

    // Stage E0[:, it*16 .. it*16+15] (interleaved) into LDS: 768 b64 elements.
#pragma unroll
    for (int l = 0; l < 3; l++) {
      int e = threadIdx.x + l * 256;        // 0..767
      int q = e >> 4;
      int c = (e & 15) * 2;
      uint32_t loff = (uint32_t)(uintptr_t)&e0s[q * 32 + c];
      async_g2l_b64(loff, E0 + (size_t)q * N2 + it * 32 + c);
    }
    async_wait0();
    __syncthreads();

    for (int q = 0; q < QD; q += 4) {
      int k0 = q + kh;
      v2f ar, ai, an, br, bi;
      ar.x = e0s[k0 * 32 + r * 2];
      ar.y = e0s[(k0 + 1) * 32 + r * 2];
      ai.x = e0s[k0 * 32 + r * 2 + 1];
      ai.y = e0s[(k0 + 1) * 32 + r * 2 + 1];
      br.x = E1[(size_t)k0 * N2 + j * 2];
      br.y = E1[(size_t)(k0 + 1) * N2 + j * 2];
      bi.x = E1[(size_t)k0 * N2 + j * 2 + 1];
      bi.y = E1[(size_t)(k0 + 1) * N2 + j * 2 + 1];
      an.x = -ai.x; an.y = -ai.y;
      accR = wmma4(ar, br, accR);   // re: +E0r^T E1r
      accR = wmma4(ai, bi, accR);   // re: +E0i^T E1i
      accI = wmma4(ar, bi, accI);   // im: +E0r^T E1i
      accI = wmma4(an, br, accI);   // im: -E0i^T E1r
    }
    // + s[b,m,h,i] * pos_enc[m,i,j]
    int col = lane & 15, rb = (lane >> 4) * 8;
    const float* Sm = ws + SOFF + (((size_t)b * MM + m) * HH + h) * NN * 2;
    for (int v = 0; v < 8; v++) {
      int ii = it * 16 + rb + v, jj = jt * 16 + col;
      float sr = Sm[2 * ii], si = Sm[2 * ii + 1];
      float pr = PR[(size_t)m * NN * NN + (size_t)ii * NN + jj];
      float pi = PI[(size_t)m * NN * NN + (size_t)ii * NN + jj];
      accR[v] += sr * pr - si * pi;
      accI[v] += sr * pi + si * pr;
    }
    __syncthreads();   // LDS reused next m
  }
  float* A = ws + AOFF + (size_t)bh * NN * NN * 2;
  int col = lane & 15, rb = (lane >> 4) * 8;
  for (int v = 0; v < 8; v++) {
    int ii = it * 16 + rb + v, jj = jt * 16 + col;
    A[((size_t)ii * NN + jj) * 2]     = accR[v];
    A[((size_t)ii * NN + jj) * 2 + 1] = accI[v];
  }
}

// ---------------------------------------------------------------------------
// Kernel 4: row softmax of |A_sum| / sqrt(48) -> real attention matrix
// ---------------------------------------------------------------------------
__global__ __launch_bounds__(256) void k_softmax(float* __restrict__ ws) {
  __shared__ float red[256];
  int row = blockIdx.x;              // (b*H + h)*N + i
  int j = threadIdx.x;
  const float* A = ws + AOFF + (size_t)row * NN * 2;
  float re = A[2 * j], im = A[2 * j + 1];
  float v = sqrtf(re * re + im * im) * 0.14433756729740643f;  // 1/sqrt(48)
  red[j] = v; __syncthreads();
  for (int s = 128; s > 0; s >>= 1) {
    if (j < s) red[j] = fmaxf(red[j], red[j + s]);
    __syncthreads();
  }
  float mx = red[0]; __syncthreads();
  float e = expf(v - mx);
  red[j] = e; __syncthreads();
  for (int s = 128; s > 0; s >>= 1) {
    if (j < s) red[j] += red[j + s];
    __syncthreads();
  }
  ws[SMOFF + (size_t)row * NN + j] = e / red[0];
}

// ---------------------------------------------------------------------------
// Kernel 5: t[b,m,h,i] = sum_j pos_enc[m,i,j] * A[b,h,i,j]
// ---------------------------------------------------------------------------
__global__ void k_t(float* __restrict__ ws) {
  int idx = blockIdx.x * blockDim.x + threadIdx.x;  // B*M*H*N
  if (idx >= BB * MM * HH * NN) return;
  int i = idx & 255; int t = idx >> 8;
  int h = t % HH; t /= HH;
  int m = t % MM; int b = t / MM;
  const float* PR = ws + POSR + (size_t)m * NN * NN + (size_t)i * NN;
  const float* PI = ws + POSI + (size_t)m * NN * NN + (size_t)i * NN;
  const float* Am = ws + SMOFF + ((size_t)(b * HH + h) * NN + i) * NN;
  float tr = 0.f, ti = 0.f;
  for (int jj = 0; jj < NN; jj++) {
    float a = Am[jj];
    tr += PR[jj] * a;
    ti += PI[jj] * a;
  }
  ws[TOFF + 2 * (size_t)idx]     = tr;
  ws[TOFF + 2 * (size_t)idx + 1] = ti;
}

// ---------------------------------------------------------------------------
// Kernel 6: flat[b,m,h*48+q,i] = sum_j V[q,j]*A[i,j] + enc1[m,h,q]*t[b,m,h,i]
// Block = 8 waves sharing one q-tile; V slab (16x512 f32 = 32KB) staged in LDS
// via async global->LDS; complex V x real A^T = 2 WMMA accs per k-step.
// ---------------------------------------------------------------------------
__global__ __launch_bounds__(256) void k_av(const float* __restrict__ enc,
                                            float* __restrict__ ws) {
  __shared__ float vs[16 * 512];                       // [ql][2j(+1)]
  int lane = threadIdx.x & 31, wave = threadIdx.x >> 5;
  int batch = blockIdx.x / 6;                 // (b,m,h): 512
  int rem   = blockIdx.x % 6;
  int qt = rem >> 1;                          // 0..2
  int it = ((rem & 1) << 3) + wave;           // 0..15
  int h = batch % HH; int t = batch / HH;
  int m = t % MM; int b = t / MM;
  int r = lane & 15, kh = (lane >> 4) * 2;

  const float* V  = ws + EOFF
      + ((((size_t)2 * BB + b) * MM + m) * HH + h) * (size_t)QD * N2
      + (size_t)qt * 16 * N2;
  const float* Am = ws + SMOFF + (size_t)(b * HH + h) * NN * NN;

  // Stage V rows qt*16..qt*16+15 (all 512 floats) into LDS: 4096 b64 elems.
#pragma unroll
  for (int l = 0; l < 16; l++) {
    int e = threadIdx.x + l * 256;            // 0..4095
    int q = e >> 8;
    int c = (e & 255) * 2;
    uint32_t loff = (uint32_t)(uintptr_t)&vs[q * 512 + c];
    async_g2l_b64(loff, V + (size_t)q * N2 + c);
  }
  async_wait0();
  __syncthreads();

  v8f accR = {}, accI = {};
  int i = it * 16 + r;   // B-fragment col (output col)
  for (int j = 0; j < NN; j += 4) {
    int k0 = j + kh;
    v2f ar, ai, bf;
    ar.x = vs[r * 512 + k0 * 2];
    ar.y = vs[r * 512 + (k0 + 1) * 2];
    ai.x = vs[r * 512 + k0 * 2 + 1];
    ai.y = vs[r * 512 + (k0 + 1) * 2 + 1];
    bf.x = Am[(size_t)i * NN + k0];           // B[k=j][n=i] = A[i,j]
    bf.y = Am[(size_t)i * NN + k0 + 1];
    accR = wmma4(ar, bf, accR);
    accI = wmma4(ai, bf, accI);
  }
  const float* e1 = enc + (((size_t)1 * MM + m) * HH + h) * QD;
  const float* T  = ws + TOFF + (((size_t)b * MM + m) * HH + h) * NN * 2;
  float* F = ws + FLATOFF + (size_t)(b * MM + m) * DD * N2
           + (size_t)(h * QD) * N2;
  int col = lane & 15, rb = (lane >> 4) * 8;
  for (int v = 0; v < 8; v++) {
    int qq = qt * 16 + rb + v, ii = it * 16 + col;
    float w = e1[qq];
    F[(size_t)qq * N2 + ii * 2]     = accR[v] + w * T[2 * ii];
    F[(size_t)qq * N2 + ii * 2 + 1] = accI[v] + w * T[2 * ii + 1];
  }
}

// ---------------------------------------------------------------------------
// Kernel 7: out[b,m] = out_w[m](384x384) @ flat[b,m]-view(384x512) -> d_out
// One wave per (b,m,mtg,nt): 3 m-tiles, B fragment loaded once.
// ---------------------------------------------------------------------------
__global__ __launch_bounds__(256) void k_out(const float* __restrict__ ow,
                                             const float* __restrict__ ws,
                                             float* __restrict__ out) {
  int lane = threadIdx.x & 31, wave = threadIdx.x >> 5;
  int grp = blockIdx.x >> 2;                  // (b,m,mtg): 512
  int nt  = ((blockIdx.x & 3) << 3) + wave;   // 0..31
  int mtg = grp & 7;                          // 0..7 -> m-tiles mtg*3..+2
  int bm  = grp >> 3;                         // 0..63
  int m = bm % MM, b = bm / MM;

  const float* A  = ow + (size_t)m * DD * DD;
  const float* Bp = ws + FLATOFF + (size_t)(b * MM + m) * DD * N2 + nt * 16;
  float* C = out + (size_t)(b * MM + m) * DD * N2 + nt * 16;

  int r = lane & 15, kh = (lane >> 4) * 2;
  v8f acc[3] = {{}, {}, {}};
  for (int k = 0; k < DD; k += 4) {
    v2f bf;
    bf.x = Bp[(size_t)(k + kh) * N2 + r];
    bf.y = Bp[(size_t)(k + kh + 1) * N2 + r];
#pragma unroll
    for (int u = 0; u < 3; u++) {
      int row = (mtg * 3 + u) * 16 + r;
      v2f a;
      a.x = A[(size_t)row * DD + k + kh];
      a.y = A[(size_t)row * DD + k + kh + 1];
      acc[u] = wmma4(a, bf, acc[u]);
    }
  }
  int col = lane & 15, rb = (lane >> 4) * 8;
#pragma unroll
  for (int u = 0; u < 3; u++)
    for (int v = 0; v < 8; v++)
      C[(size_t)((mtg * 3 + u) * 16 + rb + v) * N2 + col] = acc[u][v];
}

// ---------------------------------------------------------------------------
extern "C" void kernel_launch(void* const* d_in, const int* in_sizes, int n_in,
                              void* d_out, int out_size, void* d_ws,
                              size_t ws_size, hipStream_t stream) {
  const float* x   = (const float*)d_in[0];  // complex64 interleaved (B,M,D,N)
  const float* emb = (const float*)d_in[1];  // f32 (3,1,M,H,Q,D)
  const float* enc = (const float*)d_in[2];  // f32 (2,M,H,1,Q,1)
  const float* ow  = (const float*)d_in[3];  // f32 (M,D,H*Q)
  float* ws  = (float*)d_ws;
  float* out = (float*)d_out;                // complex64 interleaved (B,M,D,N)

  k_pos    <<<(int)((MNN + 255) / 256), 256, 0, stream>>>(ws);
  k_proj   <<<1536 * 4, 256, 0, stream>>>(x, emb, ws);
  k_s      <<<(BB * MM * HH * NN) / 256, 256, 0, stream>>>(enc, ws);
  k_attn   <<<2048, 256, 0, stream>>>(ws);
  k_softmax<<<BB * HH * NN, 256, 0, stream>>>(ws);
  k_t      <<<(BB * MM * HH * NN) / 256, 256, 0, stream>>>(ws);
  k_av     <<<512 * 6, 256, 0, stream>>>(enc, ws);
  k_out    <<<2048, 256, 0, stream>>>(ow, ws, out);
}